// VirtualNodeLayer_85109071937615
// MI455X (gfx1250) — compile-verified
//
#include <hip/hip_runtime.h>
#include <hip/hip_bf16.h>

typedef __attribute__((ext_vector_type(2))) float v2f;
typedef __attribute__((ext_vector_type(4))) float v4f;
typedef __attribute__((ext_vector_type(8))) float v8f;

#define H 256
#define HQ 64            // H/4 float4 per row
#define BGRAPHS 1024
#define BN_EPS 1e-5f

// ---------------------------------------------------------------------------
// Kernel 1: segment-sum pooling + virtual node add, float4-vectorized.
// batch_idx is sorted: binary-search the graph's [lo,hi) row range, then
// 256 threads = 4 row-groups x 64 quad-columns stream rows with b128 loads.
// ---------------------------------------------------------------------------
__device__ __forceinline__ int lower_bound_dev(const int* __restrict__ a, int n, int v) {
    int lo = 0, hi = n;
    while (lo < hi) {
        int mid = (lo + hi) >> 1;
        if (a[mid] < v) lo = mid + 1; else hi = mid;
    }
    return lo;
}

__global__ void seg_sum_kernel(const float* __restrict__ x,
                               const int* __restrict__ bidx,
                               const float* __restrict__ vnode,
                               float* __restrict__ h, int n_nodes) {
    __shared__ int s_lo, s_hi;
    __shared__ v4f red[256];
    const v4f* __restrict__ x4 = reinterpret_cast<const v4f*>(x);
    const v4f* __restrict__ v4 = reinterpret_cast<const v4f*>(vnode);
    v4f* __restrict__ h4 = reinterpret_cast<v4f*>(h);

    const int b = blockIdx.x;          // graph id
    const int t = threadIdx.x;
    const int q = t & (HQ - 1);        // quad-column 0..63
    const int g = t >> 6;              // row group 0..3
    if (t == 0) s_lo = lower_bound_dev(bidx, n_nodes, b);
    if (t == 1) s_hi = lower_bound_dev(bidx, n_nodes, b + 1);
    __syncthreads();
    const int lo = s_lo, hi = s_hi;

    v4f acc = {0.f, 0.f, 0.f, 0.f};
    for (int r = lo + g; r < hi; r += 4)
        acc += x4[(size_t)r * HQ + q];
    red[t] = acc;
    __syncthreads();
    if (t < HQ) {
        v4f s = red[t] + red[t + 64] + red[t + 128] + red[t + 192];
        h4[(size_t)b * HQ + t] = s + v4[(size_t)b * HQ + t];
    }
}

// ---------------------------------------------------------------------------
// FP32 WMMA GEMM: D[1024,256] = A[1024,256] * B[256,256] + bias, one 16x16
// output tile per wave via V_WMMA_F32_16X16X4_F32, K-loop of 64 steps.
// Per-lane operand layout (ISA 7.12.2, 32-bit A 16x4 / B 4x16 / C 16x16):
//   A: a0=A[m0+(l&15), k+2*(l>>4)],   a1=A[.., k+2*(l>>4)+1]
//   B: b0=B[k+2*(l>>4), n0+(l&15)],   b1=B[k+2*(l>>4)+1, ..]
//   D: vgpr r -> D[m0 + r + 8*(l>>4)][n0 + (l&15)]
// ---------------------------------------------------------------------------
__global__ void gemm_wmma_kernel(const float* __restrict__ A,
                                 const float* __restrict__ Bm,
                                 const float* __restrict__ bias,
                                 float* __restrict__ D) {
    const int lane = threadIdx.x & 31;
    const int wave = threadIdx.x >> 5;
    const int tile = blockIdx.x * 8 + wave;       // 1024 tiles total
    const int tm = tile >> 4;                     // 0..63
    const int tn = tile & 15;                     // 0..15
    const int m0 = tm * 16, n0 = tn * 16;
    const int r15 = lane & 15;
    const int kk = (lane >> 4) << 1;              // 0 or 2

    v8f acc = {};
    for (int k = 0; k < H; k += 4) {
        v2f a, b;
        a.x = A[(size_t)(m0 + r15) * H + k + kk];
        a.y = A[(size_t)(m0 + r15) * H + k + kk + 1];
        b.x = Bm[(size_t)(k + kk) * H + n0 + r15];
        b.y = Bm[(size_t)(k + kk + 1) * H + n0 + r15];
        acc = __builtin_amdgcn_wmma_f32_16x16x4_f32(
            false, a, false, b, (short)0, acc, false, false);
    }
    const int col = n0 + r15;
    const float bv = bias[col];
    const int mbase = m0 + ((lane >> 4) << 3);
#pragma unroll
    for (int r = 0; r < 8; ++r)
        D[(size_t)(mbase + r) * H + col] = acc[r] + bv;
}

// ---------------------------------------------------------------------------
// Kernel 3a: partial batchnorm stats.  32 blocks x 256 threads; block j
// reduces rows [32j, 32j+32) into partial sum/sumsq rows (coalesced loads,
// parallel across WGPs, fully deterministic — no float atomics).
// ---------------------------------------------------------------------------
__global__ void bn_partial_kernel(const float* __restrict__ h1,
                                  float* __restrict__ psum,
                                  float* __restrict__ psq) {
    const int j = blockIdx.x;          // 0..31
    const int c = threadIdx.x;         // feature
    float sum = 0.0f, sq = 0.0f;
    const int r0 = j * 32;
    for (int r = r0; r < r0 + 32; ++r) {
        float v = h1[(size_t)r * H + c];
        sum += v;
        sq += v * v;
    }
    psum[(size_t)j * H + c] = sum;
    psq[(size_t)j * H + c] = sq;
}

// Kernel 3b: finalize fused affine: scale = gamma*rsqrt(var+eps),
//            shift = beta - mean*scale.
__global__ void bn_finalize_kernel(const float* __restrict__ psum,
                                   const float* __restrict__ psq,
                                   const float* __restrict__ gamma,
                                   const float* __restrict__ beta,
                                   float* __restrict__ scale,
                                   float* __restrict__ shift) {
    const int c = threadIdx.x;
    float ts = 0.0f, tq = 0.0f;
    for (int j = 0; j < 32; ++j) {
        ts += psum[(size_t)j * H + c];
        tq += psq[(size_t)j * H + c];
    }
    float mean = ts * (1.0f / BGRAPHS);
    float var = tq * (1.0f / BGRAPHS) - mean * mean;
    float sc = gamma[c] * rsqrtf(var + BN_EPS);
    scale[c] = sc;
    shift[c] = beta[c] - mean * sc;
}

// ---------------------------------------------------------------------------
// Kernel 4: vn = relu(bn(h1)) @ W2 + b2.  The normalize+ReLU is fused into
// the WMMA A-operand load (per-K-element affine + max(0,.)).
// ---------------------------------------------------------------------------
__global__ void gemm_wmma_bnrelu_kernel(const float* __restrict__ h1,
                                        const float* __restrict__ W2,
                                        const float* __restrict__ b2,
                                        const float* __restrict__ scale,
                                        const float* __restrict__ shift,
                                        float* __restrict__ vn) {
    const int lane = threadIdx.x & 31;
    const int wave = threadIdx.x >> 5;
    const int tile = blockIdx.x * 8 + wave;
    const int tm = tile >> 4;
    const int tn = tile & 15;
    const int m0 = tm * 16, n0 = tn * 16;
    const int r15 = lane & 15;
    const int kk = (lane >> 4) << 1;

    v8f acc = {};
    for (int k = 0; k < H; k += 4) {
        const int k0 = k + kk, k1 = k + kk + 1;
        float a0 = h1[(size_t)(m0 + r15) * H + k0] * scale[k0] + shift[k0];
        float a1 = h1[(size_t)(m0 + r15) * H + k1] * scale[k1] + shift[k1];
        v2f a, b;
        a.x = fmaxf(a0, 0.0f);
        a.y = fmaxf(a1, 0.0f);
        b.x = W2[(size_t)k0 * H + n0 + r15];
        b.y = W2[(size_t)k1 * H + n0 + r15];
        acc = __builtin_amdgcn_wmma_f32_16x16x4_f32(
            false, a, false, b, (short)0, acc, false, false);
    }
    const int col = n0 + r15;
    const float bv = b2[col];
    const int mbase = m0 + ((lane >> 4) << 3);
#pragma unroll
    for (int r = 0; r < 8; ++r)
        vn[(size_t)(mbase + r) * H + col] = acc[r] + bv;
}

// ---------------------------------------------------------------------------
// Kernel 5: x_out = x + vn[batch_idx], float4 grid-stride pass.
// x (last read) and x_out (write-once) are 512 MB pure streams: use
// non-temporal load/store (native clang vectors) so they don't wash
// vn/weights out of L2.  vn stays RT-cached (1 MB, hit by every row).
// ---------------------------------------------------------------------------
__global__ void gather_add_kernel(const float* __restrict__ x,
                                  const int* __restrict__ bidx,
                                  const float* __restrict__ vn,
                                  float* __restrict__ xout,
                                  long n_nodes) {
    const v4f* __restrict__ x4 = reinterpret_cast<const v4f*>(x);
    const v4f* __restrict__ v4 = reinterpret_cast<const v4f*>(vn);
    v4f* __restrict__ o4 = reinterpret_cast<v4f*>(xout);
    const long nquads = n_nodes * HQ;
    const long stride = (long)gridDim.x * blockDim.x;
    for (long i = (long)blockIdx.x * blockDim.x + threadIdx.x; i < nquads; i += stride) {
        long row = i >> 6;          // HQ == 64 quads per row
        int q = (int)(i & (HQ - 1));
        int b = bidx[row];
        v4f xv = __builtin_nontemporal_load(&x4[i]);
        v4f vv = v4[(long)b * HQ + q];
        v4f r = xv + vv;
        __builtin_nontemporal_store(r, &o4[i]);
    }
}

// ---------------------------------------------------------------------------
extern "C" void kernel_launch(void* const* d_in, const int* in_sizes, int n_in,
                              void* d_out, int out_size, void* d_ws, size_t ws_size,
                              hipStream_t stream) {
    const float* x      = (const float*)d_in[0];
    const int*   bidx   = (const int*)d_in[1];
    const float* vnode  = (const float*)d_in[2];
    const float* W1     = (const float*)d_in[3];
    const float* b1     = (const float*)d_in[4];
    const float* gamma  = (const float*)d_in[5];
    const float* beta   = (const float*)d_in[6];
    const float* W2     = (const float*)d_in[7];
    const float* b2     = (const float*)d_in[8];

    const int n_nodes = in_sizes[1];           // N from batch_idx length

    float* xout = (float*)d_out;                           // [N,H]
    float* vn   = xout + (size_t)n_nodes * H;              // [B,H] (output tail)

    float* h     = (float*)d_ws;                           // [B,H]
    float* h1    = h + (size_t)BGRAPHS * H;                // [B,H]
    float* scale = h1 + (size_t)BGRAPHS * H;               // [H]
    float* shift = scale + H;                              // [H]
    float* psum  = shift + H;                              // [32,H]
    float* psq   = psum + 32 * H;                          // [32,H]

    // 1) pooled + virtual node
    seg_sum_kernel<<<BGRAPHS, 256, 0, stream>>>(x, bidx, vnode, h, n_nodes);

    // 2) h1 = h @ W1 + b1   (1024 tiles of 16x16, 8 waves/block -> 128 blocks)
    gemm_wmma_kernel<<<128, 256, 0, stream>>>(h, W1, b1, h1);

    // 3) batchnorm training stats -> fused per-feature affine (deterministic)
    bn_partial_kernel<<<32, 256, 0, stream>>>(h1, psum, psq);
    bn_finalize_kernel<<<1, 256, 0, stream>>>(psum, psq, gamma, beta, scale, shift);

    // 4) vn = relu(bn(h1)) @ W2 + b2
    gemm_wmma_bnrelu_kernel<<<128, 256, 0, stream>>>(h1, W2, b2, scale, shift, vn);

    // 5) x_out = x + vn[batch_idx]
    gather_add_kernel<<<2048, 256, 0, stream>>>(x, bidx, vn, xout, (long)n_nodes);
}